// DyHGCN_S_74148315398746
// MI455X (gfx1250) — compile-verified
//
#include <hip/hip_runtime.h>

typedef __attribute__((ext_vector_type(2))) float v2f;
typedef __attribute__((ext_vector_type(8))) float v8f;

#define TSTEPS   8
#define NEG_BIG  (-4294967295.0f)

// ---------------------------------------------------------------------------
// fp32 WMMA GEMM: C[M,N] = A[M,K] * B[K,N] (+bias, optional relu)
// One wave -> 32x64 C tile: two A fragments (rows r..r+15, r+16..r+31) share
// four B fragments -> 8x V_WMMA_F32_16X16X4_F32 per K-step. OOB rows/cols are
// handled by CLAMPING load indices (WMMA mixes neither A-rows nor B-cols, so
// clamped lanes only pollute outputs the store guard skips) -> all inner-loop
// loads are unconditional. K must be a multiple of 4 (64/72/128/512 qualify).
// ---------------------------------------------------------------------------
__global__ void wmma_gemm_f32(const float* __restrict__ A, int lda,
                              const float* __restrict__ B, int ldb,
                              float* __restrict__ C, int ldc,
                              int M, int N, int K,
                              const float* __restrict__ bias, int relu)
{
    const int lane = threadIdx.x & 31;
    const int wave = threadIdx.x >> 5;
    const int wpb  = blockDim.x >> 5;
    const int row0 = blockIdx.x * 32;
    const int col0 = (blockIdx.y * wpb + wave) * 64;   // 64-wide strip
    const int m    = lane & 15;
    const int kh   = (lane >> 4) << 1;                 // 0 or 2

    int arow0 = row0 + m;      if (arow0 > M - 1) arow0 = M - 1;
    int arow1 = row0 + 16 + m; if (arow1 > M - 1) arow1 = M - 1;
    const float* Ap0 = A + (size_t)arow0 * lda + kh;
    const float* Ap1 = A + (size_t)arow1 * lda + kh;

    const float* Bp0; const float* Bp1; const float* Bp2; const float* Bp3;
    {
        int c0 = col0 + m;      if (c0 > N - 1) c0 = N - 1;
        int c1 = col0 + 16 + m; if (c1 > N - 1) c1 = N - 1;
        int c2 = col0 + 32 + m; if (c2 > N - 1) c2 = N - 1;
        int c3 = col0 + 48 + m; if (c3 > N - 1) c3 = N - 1;
        Bp0 = B + c0; Bp1 = B + c1; Bp2 = B + c2; Bp3 = B + c3;
    }

    v8f acc00 = {}, acc01 = {}, acc02 = {}, acc03 = {};
    v8f acc10 = {}, acc11 = {}, acc12 = {}, acc13 = {};
#pragma unroll 2
    for (int k0 = 0; k0 < K; k0 += 4) {
        v2f a0 = *(const v2f*)(Ap0 + k0);              // b64 (kh, kh+1 contiguous)
        v2f a1 = *(const v2f*)(Ap1 + k0);
        size_t r0 = (size_t)(k0 + kh) * ldb;
        size_t r1 = r0 + (size_t)ldb;
        v2f b0, b1, b2, b3;
        b0.x = Bp0[r0]; b0.y = Bp0[r1];
        b1.x = Bp1[r0]; b1.y = Bp1[r1];
        b2.x = Bp2[r0]; b2.y = Bp2[r1];
        b3.x = Bp3[r0]; b3.y = Bp3[r1];
        acc00 = __builtin_amdgcn_wmma_f32_16x16x4_f32(false, a0, false, b0, (short)0, acc00, false, false);
        acc01 = __builtin_amdgcn_wmma_f32_16x16x4_f32(false, a0, false, b1, (short)0, acc01, false, false);
        acc02 = __builtin_amdgcn_wmma_f32_16x16x4_f32(false, a0, false, b2, (short)0, acc02, false, false);
        acc03 = __builtin_amdgcn_wmma_f32_16x16x4_f32(false, a0, false, b3, (short)0, acc03, false, false);
        acc10 = __builtin_amdgcn_wmma_f32_16x16x4_f32(false, a1, false, b0, (short)0, acc10, false, false);
        acc11 = __builtin_amdgcn_wmma_f32_16x16x4_f32(false, a1, false, b1, (short)0, acc11, false, false);
        acc12 = __builtin_amdgcn_wmma_f32_16x16x4_f32(false, a1, false, b2, (short)0, acc12, false, false);
        acc13 = __builtin_amdgcn_wmma_f32_16x16x4_f32(false, a1, false, b3, (short)0, acc13, false, false);
    }

    const int rhalf = (lane >> 4) << 3;                // +0 or +8
    auto store_tile = [&](const v8f& acc, int rb, int c) {
        if (c < N) {
            float bv = bias ? bias[c] : 0.0f;
#pragma unroll
            for (int i = 0; i < 8; ++i) {
                int r = rb + i;
                if (r < M) {
                    float v = acc[i] + bv;
                    if (relu) v = fmaxf(v, 0.0f);
                    C[(size_t)r * ldc + c] = v;
                }
            }
        }
    };
    store_tile(acc00, row0 + rhalf,      col0 +  0 + m);
    store_tile(acc01, row0 + rhalf,      col0 + 16 + m);
    store_tile(acc02, row0 + rhalf,      col0 + 32 + m);
    store_tile(acc03, row0 + rhalf,      col0 + 48 + m);
    store_tile(acc10, row0 + 16 + rhalf, col0 +  0 + m);
    store_tile(acc11, row0 + 16 + rhalf, col0 + 16 + m);
    store_tile(acc12, row0 + 16 + rhalf, col0 + 32 + m);
    store_tile(acc13, row0 + 16 + rhalf, col0 + 48 + m);
}

// ---------------------------------------------------------------------------
// GCN helpers
// ---------------------------------------------------------------------------
__global__ void deg_init(float* deg, int n) {
    int i = blockIdx.x * blockDim.x + threadIdx.x;
    if (i < n) deg[i] = 1.0f;               // self-loop contributes 1
}

__global__ void deg_count(const int* __restrict__ dst, int E, float* deg) {
    int i = blockIdx.x * blockDim.x + threadIdx.x;
    if (i < E) atomicAdd(&deg[dst[i]], 1.0f);
}

__global__ void deg_rsqrt(const float* __restrict__ deg, float* dinv, int n) {
    int i = blockIdx.x * blockDim.x + threadIdx.x;
    if (i < n) dinv[i] = rsqrtf(deg[i]);
}

// out[n,f] = h[n,f]*dinv[n]^2 + bias[f]   (self-loop term + bias)
__global__ void prop_init(const float* __restrict__ h, const float* __restrict__ dinv,
                          const float* __restrict__ bias, float* out,
                          int Nn, int F)
{
    size_t i = (size_t)blockIdx.x * blockDim.x + threadIdx.x;
    if (i >= (size_t)Nn * F) return;
    int n = (int)(i / F), f = (int)(i % F);
    float di = dinv[n];
    out[i] = h[i] * di * di + bias[f];
}

// One edge handled by 32 lanes; each lane does fpl features (F = 32*fpl).
__global__ void gcn_scatter(const int* __restrict__ src, const int* __restrict__ dst,
                            int E, const float* __restrict__ h,
                            const float* __restrict__ dinv, float* out,
                            int F, int fpl)
{
    long long gid = (long long)blockIdx.x * blockDim.x + threadIdx.x;
    int lane = (int)(gid & 31);
    long long e = gid >> 5;
    if (e >= E) return;
    int s = src[e], d = dst[e];
    float w = dinv[s] * dinv[d];
    const float* hp = h + (size_t)s * F + lane * fpl;
    float* op = out + (size_t)d * F + lane * fpl;
    for (int i = 0; i < fpl; ++i) atomicAdd(&op[i], hp[i] * w);
}

// ---------------------------------------------------------------------------
// dy_ts (single block): bm -> valid cumprod -> searchsorted -> repeat(STEP_LEN)
// ---------------------------------------------------------------------------
__global__ void dyts_kernel(const int* __restrict__ input_ts,
                            const int* __restrict__ graph_times,
                            int* dy_ts, int B, int L, int LP1)
{
    __shared__ int bm[64];
    int nb = L / 5;
    int j = threadIdx.x;
    if (j < nb) {
        int mx = -2147483647;
        for (int b = 0; b < B; ++b)
            for (int k = 0; k < 5; ++k) {
                int v = input_ts[b * LP1 + j * 5 + k];
                mx = v > mx ? v : mx;
            }
        bm[j] = mx;
    }
    __syncthreads();
    if (threadIdx.x == 0) {
        int valid = 1;
        for (int jj = 0; jj < nb; ++jj) {
            valid = valid && (bm[jj] >= 1);
            int cnt = 0;
            for (int t = 0; t < TSTEPS; ++t) cnt += (graph_times[t] <= bm[jj]) ? 1 : 0;
            int res = (cnt >= 1) ? (cnt - 1) : (TSTEPS - 1);
            int val = valid ? res : 0;
            for (int k = 0; k < 5; ++k) dy_ts[jj * 5 + k] = val;
        }
    }
}

// score[b,l,t] = dot64(time_emb[dy_ts[l]], dyn[t][seq[b,l]]) / (8+1e-6)
__global__ void time_score(const int* __restrict__ input, const int* __restrict__ dy_ts,
                           const float* __restrict__ time_emb, const float* __restrict__ dyn,
                           float* score, int Nn, int L, int LP1)
{
    int bl = blockIdx.x;
    int b = bl / L, l = bl % L;
    int node = input[b * LP1 + l];
    int tt = dy_ts[l];
    int d = threadIdx.x;                    // 0..31
    float te0 = time_emb[tt * 64 + d];
    float te1 = time_emb[tt * 64 + d + 32];
    for (int t = 0; t < TSTEPS; ++t) {
        const float* dp = dyn + ((size_t)t * Nn + node) * 64;
        float v = te0 * dp[d] + te1 * dp[d + 32];
        for (int o = 16; o > 0; o >>= 1) v += __shfl_down(v, o, 32);
        if (d == 0) score[(size_t)bl * TSTEPS + t] = v * (1.0f / (8.0f + 1e-6f));
    }
}

// softmax over the L axis for each (b, t)
__global__ void time_softmax(const float* __restrict__ score, float* alpha, int L)
{
    int b = blockIdx.x >> 3;
    int t = blockIdx.x & 7;
    __shared__ float red[256];
    float mx = -3.4e38f;
    for (int l = threadIdx.x; l < L; l += blockDim.x)
        mx = fmaxf(mx, score[((size_t)(b * L + l)) * TSTEPS + t]);
    red[threadIdx.x] = mx; __syncthreads();
    for (int s = 128; s > 0; s >>= 1) {
        if ((int)threadIdx.x < s) red[threadIdx.x] = fmaxf(red[threadIdx.x], red[threadIdx.x + s]);
        __syncthreads();
    }
    mx = red[0]; __syncthreads();
    float sm = 0.0f;
    for (int l = threadIdx.x; l < L; l += blockDim.x)
        sm += __expf(score[((size_t)(b * L + l)) * TSTEPS + t] - mx);
    red[threadIdx.x] = sm; __syncthreads();
    for (int s = 128; s > 0; s >>= 1) {
        if ((int)threadIdx.x < s) red[threadIdx.x] += red[threadIdx.x + s];
        __syncthreads();
    }
    float inv = 1.0f / red[0];
    for (int l = threadIdx.x; l < L; l += blockDim.x) {
        size_t idx = ((size_t)(b * L + l)) * TSTEPS + t;
        alpha[idx] = __expf(score[idx] - mx) * inv;
    }
}

// x[b,l,0:64] = sum_t alpha[b,l,t]*dyn[t][seq[b,l]] ; x[b,l,64:72] = pos_emb[l]
__global__ void build_x(const int* __restrict__ input, const float* __restrict__ alpha,
                        const float* __restrict__ dyn, const float* __restrict__ pos_emb,
                        float* x, int Nn, int L, int LP1)
{
    int bl = blockIdx.x;
    int b = bl / L, l = bl % L;
    int d = threadIdx.x;
    if (d >= 72) return;
    float v;
    if (d < 64) {
        int node = input[b * LP1 + l];
        v = 0.0f;
        for (int t = 0; t < TSTEPS; ++t)
            v += alpha[(size_t)bl * TSTEPS + t] * dyn[((size_t)t * Nn + node) * 64 + d];
    } else {
        v = pos_emb[l * 8 + (d - 64)];
    }
    x[(size_t)bl * 72 + d] = v;
}

// ---------------------------------------------------------------------------
// Transformer attention: one block per (b,h,q); 64 threads.
// Pad mask is on the QUERY position (matches reference broadcast), masked rows
// become uniform via NEG_BIG softmax.
// ---------------------------------------------------------------------------
__global__ void attention(const float* __restrict__ Q, const float* __restrict__ K,
                          const float* __restrict__ V, const int* __restrict__ input,
                          float* O, int L, int LP1)
{
    int q = blockIdx.x % L;
    int h = (blockIdx.x / L) % 8;
    int b = blockIdx.x / (L * 8);
    __shared__ float p[256];
    __shared__ float stats[1];
    int tid = threadIdx.x;
    const float* qp = Q + ((size_t)(b * L + q)) * 512 + h * 64;
    bool padq = (input[b * LP1 + q] == 0);
    for (int k = tid; k < L; k += 64) {
        float s;
        if (k > q || padq) {
            s = NEG_BIG;
        } else {
            const float* kp = K + ((size_t)(b * L + k)) * 512 + h * 64;
            s = 0.0f;
            for (int d = 0; d < 64; ++d) s += qp[d] * kp[d];
            s *= (1.0f / (8.0f + 1e-6f));
        }
        p[k] = s;
    }
    __syncthreads();
    if (tid == 0) {
        float mx = -3.4e38f;
        for (int k = 0; k < L; ++k) mx = fmaxf(mx, p[k]);
        float sm = 0.0f;
        for (int k = 0; k < L; ++k) { p[k] = __expf(p[k] - mx); sm += p[k]; }
        stats[0] = 1.0f / sm;
    }
    __syncthreads();
    float inv = stats[0];
    int d = tid;
    float a = 0.0f;
    for (int k = 0; k < L; ++k)
        a += p[k] * V[((size_t)(b * L + k)) * 512 + h * 64 + d];
    O[((size_t)(b * L + q)) * 512 + h * 64 + d] = a * inv;
}

// out = LayerNorm(a + b) over 72 dims; one wave per row.
__global__ void add_ln(const float* __restrict__ a, const float* __restrict__ b,
                       const float* __restrict__ g, const float* __restrict__ beta,
                       float* __restrict__ out)
{
    const int D = 72;
    size_t row = blockIdx.x;
    int lane = threadIdx.x;              // 0..31
    bool has2 = (lane + 64) < D;
    float x0 = a[row * D + lane]      + b[row * D + lane];
    float x1 = a[row * D + lane + 32] + b[row * D + lane + 32];
    float x2 = has2 ? (a[row * D + lane + 64] + b[row * D + lane + 64]) : 0.0f;
    float s = x0 + x1 + x2;
    for (int o = 16; o > 0; o >>= 1) s += __shfl_down(s, o, 32);
    s = __shfl(s, 0, 32);
    float mu = s / (float)D;
    float d0 = x0 - mu, d1 = x1 - mu, d2 = has2 ? (x2 - mu) : 0.0f;
    float vv = d0 * d0 + d1 * d1 + d2 * d2;
    for (int o = 16; o > 0; o >>= 1) vv += __shfl_down(vv, o, 32);
    vv = __shfl(vv, 0, 32);
    float rstd = rsqrtf(vv / (float)D + 1e-5f);
    out[row * D + lane]      = g[lane]      * d0 * rstd + beta[lane];
    out[row * D + lane + 32] = g[lane + 32] * d1 * rstd + beta[lane + 32];
    if (has2)
        out[row * D + lane + 64] = g[lane + 64] * d2 * rstd + beta[lane + 64];
}

// previous-user mask: row (b,l): token 0 and tokens seq[b, j<=l] -> -inf
__global__ void mask_kernel(const int* __restrict__ input, float* out,
                            int Nn, int L, int LP1)
{
    int bl = blockIdx.x;
    int b = bl / L, l = bl % L;
    float* row = out + (size_t)bl * Nn;
    float ninf = __int_as_float(0xff800000u);
    if (threadIdx.x == 0) row[0] = ninf;
    for (int j = threadIdx.x; j <= l; j += blockDim.x) {
        int tok = input[b * LP1 + j];
        row[tok] = ninf;
    }
}

// ---------------------------------------------------------------------------
extern "C" void kernel_launch(void* const* d_in, const int* in_sizes, int n_in,
                              void* d_out, int out_size, void* d_ws, size_t ws_size,
                              hipStream_t stream)
{
    (void)n_in; (void)out_size; (void)ws_size;
    const float* emb_g    = (const float*)d_in[0];
    const float* gcn1_W   = (const float*)d_in[1];
    const float* gcn1_b   = (const float*)d_in[2];
    const float* gcn2_W   = (const float*)d_in[3];
    const float* gcn2_b   = (const float*)d_in[4];
    const float* time_emb = (const float*)d_in[5];
    const float* pos_emb  = (const float*)d_in[6];
    const float* Wq       = (const float*)d_in[7];
    const float* Wk       = (const float*)d_in[8];
    const float* Wv       = (const float*)d_in[9];
    const float* Wo       = (const float*)d_in[10];
    const float* ln_g     = (const float*)d_in[11];
    const float* ln_b     = (const float*)d_in[12];
    const float* W1       = (const float*)d_in[13];
    const float* b1       = (const float*)d_in[14];
    const float* W2       = (const float*)d_in[15];
    const float* b2       = (const float*)d_in[16];
    const float* out_W    = (const float*)d_in[17];
    const float* out_b    = (const float*)d_in[18];
    const int*   input    = (const int*)d_in[19];
    const int*   input_ts = (const int*)d_in[20];
    const int*   edge_idx = (const int*)d_in[21];
    const int*   gtimes   = (const int*)d_in[22];

    const int Nn  = in_sizes[18];            // 50000 tokens (== out_b size)
    const int B   = 8;
    const int LP1 = in_sizes[19] / B;        // 201
    const int L   = LP1 - 1;                 // 200
    const int BL  = B * L;                   // 1600
    const int E   = in_sizes[21] / (2 * TSTEPS);

    // workspace carve-up (floats)
    float* w = (float*)d_ws;
    float* h0    = w;                          // Nn*128
    float* prop1 = h0    + (size_t)Nn * 128;   // Nn*128
    float* g2    = prop1 + (size_t)Nn * 128;   // Nn*64
    float* dyn   = g2    + (size_t)Nn * 64;    // TSTEPS*Nn*64
    float* deg   = dyn   + (size_t)TSTEPS * Nn * 64;  // Nn
    float* dinv  = deg   + Nn;                 // Nn
    float* score = dinv  + Nn;                 // BL*8
    float* alpha = score + (size_t)BL * 8;     // BL*8
    float* xbuf  = alpha + (size_t)BL * 8;     // BL*72
    float* Qb    = xbuf  + (size_t)BL * 72;    // BL*512
    float* Kb    = Qb    + (size_t)BL * 512;
    float* Vb    = Kb    + (size_t)BL * 512;
    float* attO  = Vb    + (size_t)BL * 512;   // BL*512
    float* proj  = attO  + (size_t)BL * 512;   // BL*72
    float* Xb    = proj  + (size_t)BL * 72;    // BL*72
    float* f1    = Xb    + (size_t)BL * 72;    // BL*72
    float* f2    = f1    + (size_t)BL * 72;    // BL*72
    int*   dyts  = (int*)(f2 + (size_t)BL * 72);

    auto gemm = [&](const float* A, int lda, const float* Bm, int ldb,
                    float* C, int ldc, int M, int Nc, int Kc,
                    const float* bias, int relu) {
        int strips = (Nc + 63) / 64;               // 64-col strips per row tile
        int wpb = strips < 4 ? strips : 4;         // waves per block
        dim3 grid((M + 31) / 32, (strips + wpb - 1) / wpb);
        wmma_gemm_f32<<<grid, dim3(32 * wpb), 0, stream>>>(A, lda, Bm, ldb, C, ldc,
                                                           M, Nc, Kc, bias, relu);
    };

    // ---- GCN: h0 = emb_g @ gcn1_W (timestep-invariant; compute once) ----
    gemm(emb_g, 64, gcn1_W, 128, h0, 128, Nn, 128, 64, nullptr, 0);

    for (int t = 0; t < TSTEPS; ++t) {
        const int* src = edge_idx + (size_t)t * 2 * E;
        const int* dst = src + E;
        deg_init <<<(Nn + 255) / 256, 256, 0, stream>>>(deg, Nn);
        deg_count<<<(E + 255) / 256, 256, 0, stream>>>(dst, E, deg);
        deg_rsqrt<<<(Nn + 255) / 256, 256, 0, stream>>>(deg, dinv, Nn);

        // layer 1 propagation (128 feats)
        prop_init<<<(unsigned)(((size_t)Nn * 128 + 255) / 256), 256, 0, stream>>>(
            h0, dinv, gcn1_b, prop1, Nn, 128);
        gcn_scatter<<<(unsigned)(((long long)E * 32 + 255) / 256), 256, 0, stream>>>(
            src, dst, E, h0, dinv, prop1, 128, 4);

        // layer 2: GEMM then propagation (64 feats) into dyn[t]
        gemm(prop1, 128, gcn2_W, 64, g2, 64, Nn, 64, 128, nullptr, 0);
        float* dynT = dyn + (size_t)t * Nn * 64;
        prop_init<<<(unsigned)(((size_t)Nn * 64 + 255) / 256), 256, 0, stream>>>(
            g2, dinv, gcn2_b, dynT, Nn, 64);
        gcn_scatter<<<(unsigned)(((long long)E * 32 + 255) / 256), 256, 0, stream>>>(
            src, dst, E, g2, dinv, dynT, 64, 2);
    }

    // ---- time attention over T, build x ----
    dyts_kernel <<<1, 64, 0, stream>>>(input_ts, gtimes, dyts, B, L, LP1);
    time_score  <<<BL, 32, 0, stream>>>(input, dyts, time_emb, dyn, score, Nn, L, LP1);
    time_softmax<<<B * 8, 256, 0, stream>>>(score, alpha, L);
    build_x     <<<BL, 128, 0, stream>>>(input, alpha, dyn, pos_emb, xbuf, Nn, L, LP1);

    // ---- transformer block ----
    gemm(xbuf, 72, Wq, 512, Qb, 512, BL, 512, 72, nullptr, 0);
    gemm(xbuf, 72, Wk, 512, Kb, 512, BL, 512, 72, nullptr, 0);
    gemm(xbuf, 72, Wv, 512, Vb, 512, BL, 512, 72, nullptr, 0);
    attention<<<B * 8 * L, 64, 0, stream>>>(Qb, Kb, Vb, input, attO, L, LP1);
    gemm(attO, 512, Wo, 72, proj, 72, BL, 72, 512, nullptr, 0);
    add_ln<<<BL, 32, 0, stream>>>(xbuf, proj, ln_g, ln_b, Xb);
    gemm(Xb, 72, W1, 72, f1, 72, BL, 72, 72, b1, 1);
    gemm(f1, 72, W2, 72, f2, 72, BL, 72, 72, b2, 0);
    add_ln<<<BL, 32, 0, stream>>>(Xb, f2, ln_g, ln_b, f1);   // f1 = final features

    // ---- logits + previous-user mask ----
    gemm(f1, 72, out_W, Nn, (float*)d_out, Nn, BL, Nn, 72, out_b, 0);
    mask_kernel<<<BL, 256, 0, stream>>>(input, (float*)d_out, Nn, L, LP1);
}